// HaarDecomposition2D_43456479101624
// MI455X (gfx1250) — compile-verified
//
#include <hip/hip_runtime.h>
#include <hip/hip_bf16.h>
#include <stdint.h>

// Problem constants (input (8,3,512,512) f32; output (8,30,512,512) f32).
namespace {
constexpr int kB = 8;
constexpr int kC = 3;
constexpr int kH = 512;
constexpr int kW = 512;
constexpr int kHW = kH * kW;          // 262144
constexpr int kOutC = 30;             // 9 detail levels * 3ch + 3ch low
constexpr int kTilesX = kW / 4;       // 128
constexpr int kTilesY = kH / 4;       // 128
constexpr int kTotalTiles = kB * kC * kTilesX * kTilesY; // 393216
// Zero-fill geometry: fixed grid of 2048 blocks x 256 threads = 524288 lanes,
// region per batch = 24*kHW floats = 1572864 b128 slots = exactly 3 per lane.
constexpr unsigned kZeroThreads = 2048u * 256u;
}

// True clang vector types (HIP_vector_type classes are rejected by the
// nontemporal / async-LDS builtins).
typedef float __attribute__((ext_vector_type(4))) f32x4;
typedef int   __attribute__((ext_vector_type(4))) i32x4;
typedef __attribute__((address_space(1))) i32x4 g_i32x4;  // global
typedef __attribute__((address_space(3))) i32x4 l_i32x4;  // LDS

#if __has_builtin(__builtin_amdgcn_global_store_async_from_lds_b128)
#define HAAR_HAVE_ASYNC_LDS 1
#else
#define HAAR_HAVE_ASYNC_LDS 0
#endif

// ---------------------------------------------------------------------------
// Zero-fill: channels 3..26 of every batch are exactly zero (levels 2..9
// details vanish analytically). 201 MB of streaming stores via the CDNA5
// async LDS->global path: ds_store_b128 zeroes each lane's 16B LDS slot once,
// then 24 fully-unrolled b128 async stores stream from it (ASYNCcnt-tracked).
// ---------------------------------------------------------------------------
__global__ __launch_bounds__(256) void haar_zero_kernel(float* __restrict__ out) {
  __shared__ i32x4 zbuf[256];  // pins 4 KB LDS at offset 0 (only LDS object)
  // Zero this lane's slot through a raw AS(3) pointer (opaque to DCE; same
  // address the async-store engine reads).
  l_i32x4* lslot = (l_i32x4*)(uint32_t)(threadIdx.x * (unsigned)sizeof(i32x4));
  i32x4 z = {0, 0, 0, 0};
  *lslot = z;
  asm volatile("" : : "v"(&zbuf[0]) : "memory");  // keep allocation alive
  __syncthreads();  // DScnt drained: LDS zeros visible before async stores

  unsigned tid = blockIdx.x * 256u + threadIdx.x;
  // Byte address of this lane's first slot in batch 0's zero region.
  uintptr_t p = (uintptr_t)out + (size_t)(3 * kHW) * 4 + (size_t)tid * 16;
  constexpr size_t kChunk = (size_t)kZeroThreads * 16;         // 8 MiB
  constexpr size_t kBatch = (size_t)kOutC * kHW * 4;           // 30 MiB

#pragma unroll
  for (int b = 0; b < kB; ++b) {
#pragma unroll
    for (int k = 0; k < 3; ++k) {
#if HAAR_HAVE_ASYNC_LDS
      __builtin_amdgcn_global_store_async_from_lds_b128(
          (g_i32x4*)(p + (size_t)k * kChunk), lslot, 0, 0);
#else
      f32x4 zf = {0.f, 0.f, 0.f, 0.f};
      __builtin_nontemporal_store(zf, (f32x4*)(p + (size_t)k * kChunk));
#endif
    }
    p += kBatch;
  }
#if HAAR_HAVE_ASYNC_LDS
#if __has_builtin(__builtin_amdgcn_s_wait_asynccnt)
  __builtin_amdgcn_s_wait_asynccnt(0);
#else
  asm volatile("s_wait_asynccnt 0" ::: "memory");
#endif
#endif
}

// ---------------------------------------------------------------------------
// Per-4x4-tile diagonal-Haar butterfly.
// det_level1 -> out channel c, low (== low after all 9 levels) -> ch 27+c.
// ---------------------------------------------------------------------------
__global__ __launch_bounds__(256) void haar_butterfly_kernel(
    const float* __restrict__ x, float* __restrict__ out) {
  int tid = blockIdx.x * blockDim.x + threadIdx.x;
  if (tid >= kTotalTiles) return;
  int tx = tid & (kTilesX - 1);
  int ty = (tid >> 7) & (kTilesY - 1);
  int bc = tid >> 14;
  int b = bc / kC;
  int c = bc - b * kC;

  const float* src = x + (size_t)bc * kHW + (size_t)(ty * 4) * kW + tx * 4;
  f32x4 v0 = __builtin_nontemporal_load((const f32x4*)(src + 0 * kW));
  f32x4 v1 = __builtin_nontemporal_load((const f32x4*)(src + 1 * kW));
  f32x4 v2 = __builtin_nontemporal_load((const f32x4*)(src + 2 * kW));
  f32x4 v3 = __builtin_nontemporal_load((const f32x4*)(src + 3 * kW));

  // Group g holds positions with r^c == g; top_g = x[0][g]+x[1][1^g],
  // bot_g = x[2][2^g]+x[3][3^g].
  float t0 = v0.x + v1.y, u0 = v2.z + v3.w;
  float t1 = v0.y + v1.x, u1 = v2.w + v3.z;
  float t2 = v0.z + v1.w, u2 = v2.x + v3.y;
  float t3 = v0.w + v1.z, u3 = v2.y + v3.x;

  float l0 = 0.25f * (t0 + u0), d0 = 0.25f * (t0 - u0);
  float l1 = 0.25f * (t1 + u1), d1 = 0.25f * (t1 - u1);
  float l2 = 0.25f * (t2 + u2), d2 = 0.25f * (t2 - u2);
  float l3 = 0.25f * (t3 + u3), d3 = 0.25f * (t3 - u3);

  size_t tile_off = (size_t)(ty * 4) * kW + tx * 4;
  float* det = out + (size_t)(b * kOutC + c) * kHW + tile_off;
  float* low = out + (size_t)(b * kOutC + 27 + c) * kHW + tile_off;

  // Row r gets (q[r^0], q[r^1], q[r^2], q[r^3]).
  f32x4 dr0 = {d0, d1, d2, d3}, dr1 = {d1, d0, d3, d2};
  f32x4 dr2 = {d2, d3, d0, d1}, dr3 = {d3, d2, d1, d0};
  f32x4 lr0 = {l0, l1, l2, l3}, lr1 = {l1, l0, l3, l2};
  f32x4 lr2 = {l2, l3, l0, l1}, lr3 = {l3, l2, l1, l0};
  __builtin_nontemporal_store(dr0, (f32x4*)(det + 0 * kW));
  __builtin_nontemporal_store(dr1, (f32x4*)(det + 1 * kW));
  __builtin_nontemporal_store(dr2, (f32x4*)(det + 2 * kW));
  __builtin_nontemporal_store(dr3, (f32x4*)(det + 3 * kW));
  __builtin_nontemporal_store(lr0, (f32x4*)(low + 0 * kW));
  __builtin_nontemporal_store(lr1, (f32x4*)(low + 1 * kW));
  __builtin_nontemporal_store(lr2, (f32x4*)(low + 2 * kW));
  __builtin_nontemporal_store(lr3, (f32x4*)(low + 3 * kW));
}

extern "C" void kernel_launch(void* const* d_in, const int* in_sizes, int n_in,
                              void* d_out, int out_size, void* d_ws, size_t ws_size,
                              hipStream_t stream) {
  (void)in_sizes; (void)n_in; (void)out_size; (void)d_ws; (void)ws_size;
  const float* x = (const float*)d_in[0];
  float* out = (float*)d_out;

  // Zero-fill channels 3..26 (grid geometry hardcoded in the kernel).
  hipLaunchKernelGGL(haar_zero_kernel, dim3(2048), dim3(256), 0, stream, out);
  // Butterfly: det -> ch[c], low -> ch[27+c].
  hipLaunchKernelGGL(haar_butterfly_kernel, dim3(kTotalTiles / 256), dim3(256),
                     0, stream, x, out);
}